// LSTM_25847113187333
// MI455X (gfx1250) — compile-verified
//
#include <hip/hip_runtime.h>
#include <hip/hip_bf16.h>

// Problem constants (from the reference)
#define HDIM 4
#define GATES 16      // 4*H
#define INDIM 906
#define BATCH 128
#define TSEQ 512
#define MROWS (BATCH * TSEQ)   // 65536 rows of x, flattened [B*T, IN]
#define RTILES 4               // 16-row tiles per wave (B-fragment reuse)

typedef __attribute__((ext_vector_type(2))) float v2f;
typedef __attribute__((ext_vector_type(8))) float v8f;

// ---------------------------------------------------------------------------
// Kernel 1: xp_f = x @ W_ih_f^T + (b_ih_f + b_hh_f)
// M=65536, N=16, K=906 via V_WMMA_F32_16X16X4_F32.
// One wave computes RTILES=4 vertically-stacked 16x16 tiles (64 rows),
// sharing one B (W) fragment per k-chunk across 4 WMMAs:
//   per chunk: 1 B load + 4 A loads -> 4 WMMAs  (1.25 loads/WMMA)
// and cutting W's L2 re-read traffic 4x vs one-tile-per-wave.
// A layout (16x4 f32): lanes 0-15 hold {K=k0,k0+1} of row M=lane,
//                      lanes 16-31 hold {K=k0+2,k0+3} of row M=lane-16.
// B layout mirrors A with N in the lane dimension: lane n holds
//   {B[0][n],B[1][n]} (lower half) / {B[2][n],B[3][n]} (upper half),
//   where B[k][n] = W[n][k0+k].
// ---------------------------------------------------------------------------
__global__ __launch_bounds__(32) void xproj_wmma_f32(
    const float* __restrict__ x,      // [MROWS, INDIM]
    const float* __restrict__ Wih,    // [16, INDIM]
    const float* __restrict__ bih,    // [16]
    const float* __restrict__ bhh,    // [16]
    float* __restrict__ xp)           // [MROWS, 16]
{
  const int lane = threadIdx.x & 31;
  const int half = lane >> 4;        // 0: K pair {0,1}, 1: K pair {2,3}
  const int l16  = lane & 15;
  const int row_base = blockIdx.x * (16 * RTILES);

  // row stride = 906 floats = 3624 B, divisible by 8 -> v2f loads are aligned
  const float* xrow = x   + (size_t)(row_base + l16) * INDIM + half * 2;
  const float* wrow = Wih + (size_t)l16 * INDIM + half * 2;

  v8f acc0 = {}, acc1 = {}, acc2 = {}, acc3 = {};

  // 226 full K-chunks of 4 (covers K = 0..903)
  int k = 0;
  #pragma unroll 2
  for (; k + 4 <= INDIM; k += 4) {
    v2f b  = *(const v2f*)(wrow + k);
    v2f a0 = *(const v2f*)(xrow + k);                 // rows row_base+ 0..15
    v2f a1 = *(const v2f*)(xrow + k + 1 * 16 * INDIM); // rows +16..31
    v2f a2 = *(const v2f*)(xrow + k + 2 * 16 * INDIM); // rows +32..47
    v2f a3 = *(const v2f*)(xrow + k + 3 * 16 * INDIM); // rows +48..63
    acc0 = __builtin_amdgcn_wmma_f32_16x16x4_f32(false, a0, false, b, (short)0, acc0, false, false);
    acc1 = __builtin_amdgcn_wmma_f32_16x16x4_f32(false, a1, false, b, (short)0, acc1, false, false);
    acc2 = __builtin_amdgcn_wmma_f32_16x16x4_f32(false, a2, false, b, (short)0, acc2, false, false);
    acc3 = __builtin_amdgcn_wmma_f32_16x16x4_f32(false, a3, false, b, (short)0, acc3, false, false);
  }

  // tail: K = 904,905 valid only for the lower half's K-pair
  {
    v2f b  = {0.f, 0.f};
    v2f a0 = {0.f, 0.f}, a1 = {0.f, 0.f}, a2 = {0.f, 0.f}, a3 = {0.f, 0.f};
    if (half == 0) {               // reconverges before the WMMAs (EXEC all-1s)
      b  = *(const v2f*)(wrow + 904);
      a0 = *(const v2f*)(xrow + 904);
      a1 = *(const v2f*)(xrow + 904 + 1 * 16 * INDIM);
      a2 = *(const v2f*)(xrow + 904 + 2 * 16 * INDIM);
      a3 = *(const v2f*)(xrow + 904 + 3 * 16 * INDIM);
    }
    acc0 = __builtin_amdgcn_wmma_f32_16x16x4_f32(false, a0, false, b, (short)0, acc0, false, false);
    acc1 = __builtin_amdgcn_wmma_f32_16x16x4_f32(false, a1, false, b, (short)0, acc1, false, false);
    acc2 = __builtin_amdgcn_wmma_f32_16x16x4_f32(false, a2, false, b, (short)0, acc2, false, false);
    acc3 = __builtin_amdgcn_wmma_f32_16x16x4_f32(false, a3, false, b, (short)0, acc3, false, false);
  }

  // C/D layout: VGPR j -> M = tile_base + j + half*8, N = lane&15
  const float bias = bih[l16] + bhh[l16];
  float* orow = xp + ((size_t)(row_base + half * 8)) * GATES + l16;
  #pragma unroll
  for (int j = 0; j < 8; ++j) {
    orow[(size_t)(j + 0 * 16) * GATES] = acc0[j] + bias;
    orow[(size_t)(j + 1 * 16) * GATES] = acc1[j] + bias;
    orow[(size_t)(j + 2 * 16) * GATES] = acc2[j] + bias;
    orow[(size_t)(j + 3 * 16) * GATES] = acc3[j] + bias;
  }
}

// ---------------------------------------------------------------------------
// Kernel 2: fused (a) backward LSTM single step on x[:, T-1]
//                 (b) forward LSTM scan over T=512
//                 (c) final [8 -> 2] projection.
// 16 lanes per batch element (one lane per gate), 2 batches per wave.
// Gate order (torch): i=0..3, f=4..7, g=8..11, o=12..15.
// ---------------------------------------------------------------------------
__device__ __forceinline__ float fast_sigmoid(float v) {
  return __builtin_amdgcn_rcpf(1.0f + __expf(-v));
}
__device__ __forceinline__ float fast_tanh(float v) {
  return 1.0f - 2.0f * __builtin_amdgcn_rcpf(1.0f + __expf(2.0f * v));
}

__global__ __launch_bounds__(256) void lstm_scan_fused(
    const float* __restrict__ xp,     // [B, T, 16] forward pre-activations
    const float* __restrict__ Whh_f,  // [16, 4]
    const float* __restrict__ x,      // [B, T, INDIM]
    const float* __restrict__ Wih_b,  // [16, INDIM]
    const float* __restrict__ bih_b,  // [16]
    const float* __restrict__ bhh_b,  // [16]
    const float* __restrict__ Wout,   // [2, 8]
    const float* __restrict__ bout,   // [2]
    float* __restrict__ out)          // [B, 2]
{
  const int tid  = blockIdx.x * blockDim.x + threadIdx.x;  // 0..2047
  const int b    = tid >> 4;         // batch element
  const int g    = tid & 15;         // gate index
  const int lane = threadIdx.x & 31;
  const int grpbase = lane & 16;     // base lane of this batch's 16-lane group
  const bool isG = (g >= 8) && (g < 12);   // tanh gate, else sigmoid

  // ---- (a) backward direction: single cell step from h0=c0=0 -------------
  // preact = x[b, T-1, :] . W_ih_b[g, :] + b_ih_b[g] + b_hh_b[g]
  const float* xr = x     + ((size_t)b * TSEQ + (TSEQ - 1)) * INDIM;
  const float* wr = Wih_b + (size_t)g * INDIM;
  float pb = bih_b[g] + bhh_b[g];
  #pragma unroll 4
  for (int k = 0; k < INDIM; k += 2) {   // rows are 8B aligned
    float2 xv = *(const float2*)(xr + k);
    float2 wv = *(const float2*)(wr + k);
    pb += xv.x * wv.x + xv.y * wv.y;
  }
  float ab = isG ? fast_tanh(pb) : fast_sigmoid(pb);
  // unit lane u (group lane u<4): i=own, g=lane^8, o=lane^12 (f*c0 = 0)
  float gb = __shfl_xor(ab, 8);
  float ob = __shfl_xor(ab, 12);
  float cb = ab * gb;                  // c = i * g   (f*c0 vanishes)
  float hbv = ob * fast_tanh(cb);      // h_b (valid on unit lanes)
  const float hb0 = __shfl(hbv, grpbase + 0);
  const float hb1 = __shfl(hbv, grpbase + 1);
  const float hb2 = __shfl(hbv, grpbase + 2);
  const float hb3 = __shfl(hbv, grpbase + 3);

  // ---- (b) forward scan over T=512 ---------------------------------------
  const float w0 = Whh_f[g * HDIM + 0];
  const float w1 = Whh_f[g * HDIM + 1];
  const float w2 = Whh_f[g * HDIM + 2];
  const float w3 = Whh_f[g * HDIM + 3];

  const float* xpf = xp + (size_t)b * TSEQ * GATES + g;

  float h0 = 0.f, h1 = 0.f, h2 = 0.f, h3 = 0.f;
  float c  = 0.f;

  float pre_n = xpf[0];                // software-pipelined xp load
  for (int t = 0; t < TSEQ; ++t) {
    const float pre = pre_n;
    const int tn = (t + 1 < TSEQ) ? (t + 1) : t;
    pre_n = xpf[(size_t)tn * GATES];   // issue next load before the math

    const float p = pre + w0 * h0 + w1 * h1 + w2 * h2 + w3 * h3;
    const float a = isG ? fast_tanh(p) : fast_sigmoid(p);

    // gather i/f/g/o onto unit lanes u = lane&3
    const float fa = __shfl_xor(a, 4);
    const float ga = __shfl_xor(a, 8);
    const float oa = __shfl_xor(a, 12);

    const float cn = fa * c + a * ga;      // valid on unit lanes
    const float hn = oa * fast_tanh(cn);
    c = cn;

    // broadcast new h[0..3] to every lane of the group
    h0 = __shfl(hn, grpbase + 0);
    h1 = __shfl(hn, grpbase + 1);
    h2 = __shfl(hn, grpbase + 2);
    h3 = __shfl(hn, grpbase + 3);
  }

  // ---- (c) out[b] = concat(h_f, h_b) @ W_out^T + b_out -------------------
  if (g < 2) {
    const float* wo = Wout + g * 8;
    float r = bout[g]
            + wo[0] * h0 + wo[1] * h1 + wo[2] * h2 + wo[3] * h3
            + wo[4] * hb0 + wo[5] * hb1 + wo[6] * hb2 + wo[7] * hb3;
    out[b * 2 + g] = r;
  }
}

// ---------------------------------------------------------------------------
extern "C" void kernel_launch(void* const* d_in, const int* in_sizes, int n_in,
                              void* d_out, int out_size, void* d_ws, size_t ws_size,
                              hipStream_t stream) {
  const float* x      = (const float*)d_in[0];
  const float* W_ih_f = (const float*)d_in[1];
  const float* W_hh_f = (const float*)d_in[2];
  const float* b_ih_f = (const float*)d_in[3];
  const float* b_hh_f = (const float*)d_in[4];
  const float* W_ih_b = (const float*)d_in[5];
  // d_in[6] = W_hh_b: unused (h0 = 0 for the single backward step)
  const float* b_ih_b = (const float*)d_in[7];
  const float* b_hh_b = (const float*)d_in[8];
  const float* W_out  = (const float*)d_in[9];
  const float* b_out  = (const float*)d_in[10];
  float* out = (float*)d_out;

  float* xp = (float*)d_ws;            // 65536 * 16 floats = 4 MB

  // 1024 waves, each computing 64 rows x 16 gates
  xproj_wmma_f32<<<dim3(MROWS / (16 * RTILES)), dim3(32), 0, stream>>>(
      x, W_ih_f, b_ih_f, b_hh_f, xp);

  // 128 batches * 16 gate-lanes = 2048 threads
  lstm_scan_fused<<<dim3((BATCH * GATES) / 256), dim3(256), 0, stream>>>(
      xp, W_hh_f, x, W_ih_b, b_ih_b, b_hh_b, W_out, b_out, out);

  (void)in_sizes; (void)n_in; (void)out_size; (void)ws_size;
}